// NccLossBak_50216757625543
// MI455X (gfx1250) — compile-verified
//
#include <hip/hip_runtime.h>
#include <stdint.h>

// NCC loss, fused 3D box-filter tile kernel for gfx1250 (MI455X).
// Box window 9x9x9, volumes [2,1,160,160,160] fp32.

#define WIN   9
#define PAD   4                 // WIN/2
#define TILE  16
#define HALO  (TILE + WIN - 1)  // 24
#define NVOX_TILE (HALO*HALO*HALO)      // 13824
#define NX    160
#define NTILE (NX / TILE)       // 10
#define NBLK  (2 * NTILE * NTILE * NTILE) // 2000

typedef __attribute__((ext_vector_type(2))) float v2f;
typedef __attribute__((ext_vector_type(8))) float v8f;

// ---- CDNA5 async global->LDS copy (ASYNCcnt path) ------------------------
__device__ __forceinline__ void async_load_f32(uint32_t lds_byte_off, uint64_t gaddr) {
    // GV mode: vaddr = 64-bit VGPR pair, no saddr. VDST VGPR = LDS byte offset.
    asm volatile("global_load_async_to_lds_b32 %0, %1, off"
                 :: "v"(lds_byte_off), "v"(gaddr) : "memory");
}
__device__ __forceinline__ void wait_async_zero() {
    asm volatile("s_wait_asynccnt 0" ::: "memory");
}

// --------------------------------------------------------------------------
// One block = one 16^3 output tile. LDS holds the 24^3 halo tiles of I and J,
// plus W-summed (24x24x16) and WH-summed (24x16x16) scratch.
// D-axis 9-tap sum done as Band(16x24) x T(24x16) via v_wmma_f32_16x16x4_f32.
// --------------------------------------------------------------------------
extern "C" __global__ void __launch_bounds__(256)
ncc_tile_kernel(const float* __restrict__ I, const float* __restrict__ J,
                float* __restrict__ partials) {
    extern __shared__ float smem[];
    float* sI   = smem;                       // 13824 floats
    float* sJ   = sI + NVOX_TILE;             // 13824
    float* tmp1 = sJ + NVOX_TILE;             // 24*24*16 = 9216 (W-summed)
    float* tmp2 = tmp1 + (HALO*HALO*TILE);    // 24*16*16 = 6144 (WH-summed)
    float* red  = tmp2 + (HALO*TILE*TILE);    // 8 (wave partials)

    const int tid = threadIdx.x;
    const int blk = blockIdx.x;
    const int b   = blk / (NTILE*NTILE*NTILE);
    int t         = blk % (NTILE*NTILE*NTILE);
    const int tz  = t / (NTILE*NTILE);
    const int ty  = (t / NTILE) % NTILE;
    const int tx  = t % NTILE;
    const int z0  = tz*TILE - PAD, y0 = ty*TILE - PAD, x0 = tx*TILE - PAD;

    // ---- Stage 1: zero-pad (edge tiles only), then async-load halo tiles ----
    const bool edge = (tx==0)|(ty==0)|(tz==0)|(tx==NTILE-1)|(ty==NTILE-1)|(tz==NTILE-1);
    if (edge) {
        for (int e = tid; e < 2*NVOX_TILE; e += 256) smem[e] = 0.0f;
    }
    __syncthreads();  // ds stores complete before async loads write same LDS

    const uint64_t baseI = (uint64_t)(uintptr_t)I + (uint64_t)b * (NX*NX*NX) * 4ull;
    const uint64_t baseJ = (uint64_t)(uintptr_t)J + (uint64_t)b * (NX*NX*NX) * 4ull;
    const uint32_t sI_off = (uint32_t)(uintptr_t)sI;   // LDS byte offset (addr[31:0])
    const uint32_t sJ_off = (uint32_t)(uintptr_t)sJ;

    for (int e = tid; e < NVOX_TILE; e += 256) {       // 13824/256 = 54, uniform
        const int x = e % HALO;
        const int y = (e / HALO) % HALO;
        const int z = e / (HALO*HALO);
        const int gx = x0 + x, gy = y0 + y, gz = z0 + z;
        const bool in = (gx>=0)&(gx<NX)&(gy>=0)&(gy<NX)&(gz>=0)&(gz<NX);
        if (in) {
            const uint64_t goff = 4ull * (uint64_t)((gz*NX + gy)*NX + gx);
            async_load_f32(sI_off + 4u*(uint32_t)e, baseI + goff);
            async_load_f32(sJ_off + 4u*(uint32_t)e, baseJ + goff);
        }
    }
    wait_async_zero();
    __syncthreads();

    // ---- Stage 2: five fields, separable box sums, D-axis via WMMA ----
    const int lane = tid & 31;
    const int wv   = tid >> 5;          // 8 waves
    const int half = lane >> 4;         // 0: K 0-1 of each step / M rows 0-7, 1: K 2-3 of each step / M rows 8-15
    const int nn   = lane & 15;         // column within group / A-matrix row M

    v8f acc[5][2];
    #pragma unroll
    for (int f = 0; f < 5; ++f)
        #pragma unroll
        for (int gi = 0; gi < 2; ++gi)
            #pragma unroll
            for (int r = 0; r < 8; ++r) acc[f][gi][r] = 0.0f;

    #pragma unroll
    for (int f = 0; f < 5; ++f) {
        // field value from the raw tiles (f: 0=I 1=J 2=I2 3=J2 4=IJ)
        auto FLD = [&](int idx) -> float {
            if (f == 0) return sI[idx];
            if (f == 1) return sJ[idx];
            if (f == 2) { float a = sI[idx]; return a*a; }
            if (f == 3) { float a = sJ[idx]; return a*a; }
            return sI[idx] * sJ[idx];
        };

        // W-axis running 9-tap sum: 24x24 rows -> 16 outputs each
        for (int row = tid; row < HALO*HALO; row += 256) {
            const int bin  = row * HALO;
            const int bout = row * TILE;
            float s = 0.0f;
            #pragma unroll
            for (int k = 0; k < WIN; ++k) s += FLD(bin + k);
            tmp1[bout] = s;
            for (int x = 1; x < TILE; ++x) {
                s += FLD(bin + x + WIN - 1) - FLD(bin + x - 1);
                tmp1[bout + x] = s;
            }
        }
        __syncthreads();

        // H-axis running 9-tap sum: 24(z) x 16(x') columns -> 16 outputs each
        for (int col = tid; col < HALO*TILE; col += 256) {
            const int z  = col / TILE;
            const int xp = col % TILE;
            const int in0  = z*HALO*TILE + xp;  // tmp1[(z*24 + y)*16 + xp]
            const int out0 = z*TILE*TILE + xp;  // tmp2[(z*16 + yp)*16 + xp]
            float s = 0.0f;
            #pragma unroll
            for (int k = 0; k < WIN; ++k) s += tmp1[in0 + k*TILE];
            tmp2[out0] = s;
            for (int yp = 1; yp < TILE; ++yp) {
                s += tmp1[in0 + (yp + WIN - 1)*TILE] - tmp1[in0 + (yp - 1)*TILE];
                tmp2[out0 + yp*TILE] = s;
            }
        }
        __syncthreads();

        // D-axis 9-tap sum as WMMA: Out(16x16) = Band(16x24) x tmp2(24x16cols)
        // group g == y'; columns of the group are x' = 0..15.
        #pragma unroll
        for (int gi = 0; gi < 2; ++gi) {
            const int g     = wv*2 + gi;
            const int cbase = g*16 + nn;        // flat column (y'*16 + x')
            v8f c = acc[f][gi];
            #pragma unroll
            for (int s4 = 0; s4 < 6; ++s4) {    // K = 24 in 6 steps of 4
                const int kk0 = 4*s4 + 2*half;  // this lane's K for VGPR0
                const int kk1 = kk0 + 1;        // ... for VGPR1
                v2f a, bm;
                // Band: A[m][kk] = 1 iff m <= kk <= m+8  (m = nn)
                a.x = (kk0 >= nn && kk0 < nn + WIN) ? 1.0f : 0.0f;
                a.y = (kk1 >= nn && kk1 < nn + WIN) ? 1.0f : 0.0f;
                // B[kk][n] = tmp2[kk*256 + cbase]
                bm.x = tmp2[kk0*256 + cbase];
                bm.y = tmp2[kk1*256 + cbase];
                c = __builtin_amdgcn_wmma_f32_16x16x4_f32(
                        /*neg_a=*/false, a, /*neg_b=*/false, bm,
                        /*c_mod=*/(short)0, c, /*reuse_a=*/false, /*reuse_b=*/false);
            }
            acc[f][gi] = c;
        }
        __syncthreads();  // tmp1/tmp2 reused by next field
    }

    // ---- Stage 3: per-voxel cc in registers, block reduce ----
    // Lane holds voxels (z' = r + 8*half, y' = g, x' = nn), r = 0..7, gi = 0..1.
    const float wsz  = 729.0f;
    const float winv = 1.0f / 729.0f;
    float ccsum = 0.0f;
    #pragma unroll
    for (int gi = 0; gi < 2; ++gi) {
        #pragma unroll
        for (int r = 0; r < 8; ++r) {
            const float SI  = acc[0][gi][r];
            const float SJ  = acc[1][gi][r];
            const float SI2 = acc[2][gi][r];
            const float SJ2 = acc[3][gi][r];
            const float SIJ = acc[4][gi][r];
            const float uI = SI * winv;
            const float uJ = SJ * winv;
            const float cross = SIJ - uJ*SI - uI*SJ + uI*uJ*wsz;
            const float Ivar  = SI2 - 2.0f*uI*SI + uI*uI*wsz;
            const float Jvar  = SJ2 - 2.0f*uJ*SJ + uJ*uJ*wsz;
            ccsum += cross * cross / (Ivar * Jvar + 1e-5f);
        }
    }
    // wave32 butterfly reduce (deterministic)
    #pragma unroll
    for (int o = 16; o > 0; o >>= 1) ccsum += __shfl_xor(ccsum, o, 32);
    if (lane == 0) red[wv] = ccsum;
    __syncthreads();
    if (tid == 0) {
        float s = 0.0f;
        #pragma unroll
        for (int i = 0; i < 8; ++i) s += red[i];
        partials[blk] = s;   // overwrite (no atomics) -> bitwise deterministic
    }
}

// Deterministic fixed-order final reduction: loss = 1 - sum(cc)/N
extern "C" __global__ void __launch_bounds__(256)
ncc_finalize(const float* __restrict__ partials, int n, float* __restrict__ out) {
    __shared__ float red[256];
    float s = 0.0f;
    for (int i = threadIdx.x; i < n; i += 256) s += partials[i];
    red[threadIdx.x] = s;
    __syncthreads();
    for (int o = 128; o > 0; o >>= 1) {
        if ((int)threadIdx.x < o) red[threadIdx.x] += red[threadIdx.x + o];
        __syncthreads();
    }
    if (threadIdx.x == 0)
        out[0] = 1.0f - red[0] / (float)(2 * NX * NX * NX);
}

extern "C" void kernel_launch(void* const* d_in, const int* in_sizes, int n_in,
                              void* d_out, int out_size, void* d_ws, size_t ws_size,
                              hipStream_t stream) {
    (void)in_sizes; (void)n_in; (void)out_size; (void)ws_size;
    const float* I = (const float*)d_in[0];   // y_true
    const float* J = (const float*)d_in[1];   // y_pred
    float* partials = (float*)d_ws;           // NBLK floats of scratch

    const size_t shmem =
        (size_t)(2*NVOX_TILE + HALO*HALO*TILE + HALO*TILE*TILE + 8) * sizeof(float);

    ncc_tile_kernel<<<NBLK, 256, shmem, stream>>>(I, J, partials);
    ncc_finalize<<<1, 256, 0, stream>>>(partials, NBLK, (float*)d_out);
}